// SAGPoolNet_8778913153585
// MI455X (gfx1250) — compile-verified
//
#include <hip/hip_runtime.h>
#include <math.h>

typedef __attribute__((ext_vector_type(2))) float v2f;
typedef __attribute__((ext_vector_type(8))) float v8f;

#define FIN    16
#define NHID   128
#define NPG    1024
#define KEEP   512
#define NGRAPH 128

// ---------------- degree / normalization ----------------
__global__ void deg_count_kernel(const int* __restrict__ dst, int* __restrict__ cnt, int E) {
  int e = blockIdx.x * blockDim.x + threadIdx.x;
  if (e < E) atomicAdd(&cnt[dst[e]], 1);
}

// in-place: buf holds int counts on entry, float rsqrt(1+deg) on exit
__global__ void dis_kernel(void* buf, int Nn) {
  int i = blockIdx.x * blockDim.x + threadIdx.x;
  if (i < Nn) {
    int c = ((int*)buf)[i];
    ((float*)buf)[i] = rsqrtf(1.0f + (float)c);
  }
}

// ---------------- f32 WMMA GEMM: C[M x Nc] = A[M x Kd] @ B[Kd x Nc] (+bias, relu) ----------------
// One wave computes one 16x16 tile via V_WMMA_F32_16X16X4_F32, K stepped by 4.
// A layout (16x4): lanes 0-15 -> row=lane, K={k,k+1}; lanes 16-31 -> row=lane-16, K={k+2,k+3}
// B layout (4x16): lanes 0-15 -> col=lane, K={k,k+1}; lanes 16-31 -> col=lane-16, K={k+2,k+3}
// D layout (16x16): vgpr v: lanes 0-15 -> row v, lanes 16-31 -> row v+8; col = lane&15
__global__ void wmma_gemm_f32_kernel(const float* __restrict__ A,
                                     const float* __restrict__ Bm,
                                     const float* __restrict__ bias,
                                     float* __restrict__ C,
                                     int M, int Kd, int Nc, int do_relu) {
  int wid  = (blockIdx.x * blockDim.x + threadIdx.x) >> 5;
  int lane = threadIdx.x & 31;
  int tilesN = Nc >> 4;
  int tiles  = (M >> 4) * tilesN;
  if (wid >= tiles) return;                 // wave-uniform branch; surviving waves have EXEC all-ones
  int tm = wid / tilesN, tn = wid - tm * tilesN;
  int m0 = tm << 4, n0 = tn << 4;
  int l16   = lane & 15;
  int khalf = lane >> 4;                    // 0 or 1 -> K offset 0 or 2

  const float* arow = A  + (size_t)(m0 + l16) * Kd + 2 * khalf;
  const float* bcol = Bm + (size_t)(2 * khalf) * Nc + n0 + l16;

  v8f acc = {};
  for (int k = 0; k < Kd; k += 4) {
    v2f a = *(const v2f*)(arow + k);        // A[row][k+2*khalf], A[row][k+2*khalf+1]
    v2f b;
    b.x = bcol[(size_t)k * Nc];             // B[k+2*khalf  ][col]
    b.y = bcol[(size_t)(k + 1) * Nc];       // B[k+2*khalf+1][col]
    acc = __builtin_amdgcn_wmma_f32_16x16x4_f32(false, a, false, b,
                                                (short)0, acc, false, false);
  }

  int col = n0 + l16;
  float bv = bias ? bias[col] : 0.0f;
  int rbase = m0 + khalf * 8;
#pragma unroll
  for (int v = 0; v < 8; ++v) {
    float val = acc[v] + bv;
    if (do_relu) val = fmaxf(val, 0.0f);
    C[(size_t)(rbase + v) * Nc + col] = val;
  }
}

// ---------------- edge scatter: agg[dst] += dis[src]*dis[dst] * xw[src]  (one wave per edge) ----------------
__global__ void scatter_kernel(const int* __restrict__ src, const int* __restrict__ dst,
                               const float* __restrict__ dis, const float* __restrict__ xw,
                               float* __restrict__ agg, int E) {
  int t = blockIdx.x * blockDim.x + threadIdx.x;
  int e = t >> 5;
  int lane = t & 31;
  if (e >= E) return;
  int s = src[e], d = dst[e];
  float nrm = dis[s] * dis[d];
  float4 v = ((const float4*)(xw + (size_t)s * NHID))[lane];
  float* ad = agg + (size_t)d * NHID + lane * 4;
  unsafeAtomicAdd(ad + 0, nrm * v.x);
  unsafeAtomicAdd(ad + 1, nrm * v.y);
  unsafeAtomicAdd(ad + 2, nrm * v.z);
  unsafeAtomicAdd(ad + 3, nrm * v.w);
}

// h = relu(agg + dis^2 * xw + b), in place on h (holds agg on entry)
__global__ void combine_kernel(float* __restrict__ h, const float* __restrict__ xw,
                               const float* __restrict__ dis, const float* __restrict__ bias,
                               int Nn) {
  int t = blockIdx.x * blockDim.x + threadIdx.x;
  if (t >= Nn * NHID) return;
  int i = t >> 7, f = t & (NHID - 1);
  float dv = dis[i];
  float val = h[t] + dv * dv * xw[t] + bias[f];
  h[t] = fmaxf(val, 0.0f);
}

// ---------------- score GCN (3*NHID -> 1) ----------------
__global__ void score_linear_kernel(const float* __restrict__ h1, const float* __restrict__ h2,
                                    const float* __restrict__ h3, const float* __restrict__ Wp,
                                    float* __restrict__ xwp, int Nn) {
  int i = blockIdx.x * blockDim.x + threadIdx.x;
  if (i >= Nn) return;
  const float* r1 = h1 + (size_t)i * NHID;
  const float* r2 = h2 + (size_t)i * NHID;
  const float* r3 = h3 + (size_t)i * NHID;
  float s = 0.0f;
  for (int k = 0; k < NHID; ++k) s += r1[k] * Wp[k];
  for (int k = 0; k < NHID; ++k) s += r2[k] * Wp[NHID + k];
  for (int k = 0; k < NHID; ++k) s += r3[k] * Wp[2 * NHID + k];
  xwp[i] = s;
}

__global__ void scatter_score_kernel(const int* __restrict__ src, const int* __restrict__ dst,
                                     const float* __restrict__ dis, const float* __restrict__ xwp,
                                     float* __restrict__ sc, int E) {
  int e = blockIdx.x * blockDim.x + threadIdx.x;
  if (e >= E) return;
  int s = src[e], d = dst[e];
  unsafeAtomicAdd(&sc[d], dis[s] * dis[d] * xwp[s]);
}

__global__ void score_combine_kernel(float* __restrict__ sc, const float* __restrict__ xwp,
                                     const float* __restrict__ dis, const float* __restrict__ bp,
                                     int Nn) {
  int i = blockIdx.x * blockDim.x + threadIdx.x;
  if (i >= Nn) return;
  float dv = dis[i];
  sc[i] = sc[i] + dv * dv * xwp[i] + bp[0];
}

// ---------------- per-graph top-K via LDS bitonic sort (desc by value, ties -> lower index) ----------------
__global__ __launch_bounds__(NPG) void topk_kernel(const float* __restrict__ score,
                                                   int* __restrict__ topi,
                                                   float* __restrict__ topgate) {
  __shared__ float sv[NPG];
  __shared__ int   si[NPG];
  int g = blockIdx.x, t = threadIdx.x;
  sv[t] = score[(size_t)g * NPG + t];
  si[t] = t;
  __syncthreads();
  for (int k = 2; k <= NPG; k <<= 1) {
    for (int j = k >> 1; j > 0; j >>= 1) {
      int ixj = t ^ j;
      if (ixj > t) {
        float va = sv[t], vb = sv[ixj];
        int   ia = si[t], ib = si[ixj];
        bool aBeforeB = (va > vb) || (va == vb && ia < ib);  // "descending" order
        bool desc = ((t & k) == 0);
        if (desc ? !aBeforeB : aBeforeB) {
          sv[t] = vb; sv[ixj] = va;
          si[t] = ib; si[ixj] = ia;
        }
      }
      __syncthreads();
    }
  }
  if (t < KEEP) {
    topi[g * KEEP + t]    = g * NPG + si[t];   // global node id
    topgate[g * KEEP + t] = tanhf(sv[t]);
  }
}

// ---------------- gated max/mean pool -> g [NGRAPH x 768] ----------------
__global__ void pool_kernel(const float* __restrict__ h1, const float* __restrict__ h2,
                            const float* __restrict__ h3, const int* __restrict__ topi,
                            const float* __restrict__ topgate, float* __restrict__ gout) {
  int g = blockIdx.x, f = threadIdx.x;      // f in [0, 384)
  const float* hs = (f < NHID) ? h1 : (f < 2 * NHID) ? h2 : h3;
  int c = f & (NHID - 1);
  float mx = -INFINITY, sm = 0.0f;
  for (int j = 0; j < KEEP; ++j) {
    int node = topi[g * KEEP + j];
    float gate = topgate[g * KEEP + j];
    float v = hs[(size_t)node * NHID + c] * gate;
    mx = fmaxf(mx, v);
    sm += v;
  }
  gout[g * 768 + f]       = mx;             // gmp
  gout[g * 768 + 384 + f] = sm * (1.0f / KEEP);  // gap
}

// ---------------- final 64->2 + log_softmax ----------------
__global__ void head_kernel(const float* __restrict__ gin, const float* __restrict__ WL3,
                            const float* __restrict__ bL3, float* __restrict__ out) {
  int g = blockIdx.x * blockDim.x + threadIdx.x;
  if (g >= NGRAPH) return;
  float z0 = bL3[0], z1 = bL3[1];
  for (int k = 0; k < 64; ++k) {
    float v = gin[g * 64 + k];
    z0 += v * WL3[2 * k];
    z1 += v * WL3[2 * k + 1];
  }
  float m = fmaxf(z0, z1);
  float lse = m + logf(expf(z0 - m) + expf(z1 - m));
  out[2 * g]     = z0 - lse;
  out[2 * g + 1] = z1 - lse;
}

// ---------------- host orchestration ----------------
extern "C" void kernel_launch(void* const* d_in, const int* in_sizes, int n_in,
                              void* d_out, int out_size, void* d_ws, size_t ws_size,
                              hipStream_t stream) {
  const float* x   = (const float*)d_in[0];
  const int*   ei  = (const int*)d_in[1];
  const float* W1  = (const float*)d_in[2];  const float* b1  = (const float*)d_in[3];
  const float* W2  = (const float*)d_in[4];  const float* b2  = (const float*)d_in[5];
  const float* W3  = (const float*)d_in[6];  const float* b3  = (const float*)d_in[7];
  const float* Wp  = (const float*)d_in[8];  const float* bp  = (const float*)d_in[9];
  const float* WL1 = (const float*)d_in[10]; const float* bL1 = (const float*)d_in[11];
  const float* WL2 = (const float*)d_in[12]; const float* bL2 = (const float*)d_in[13];
  const float* WL3 = (const float*)d_in[14]; const float* bL3 = (const float*)d_in[15];

  const int Nn = in_sizes[0] / FIN;         // 131072
  const int E  = in_sizes[1] / 2;           // 1048576
  const int* src = ei;
  const int* dst = ei + E;

  char* ws = (char*)d_ws;
  float* h1  = (float*)ws; ws += (size_t)Nn * NHID * 4;
  float* h2  = (float*)ws; ws += (size_t)Nn * NHID * 4;
  float* h3  = (float*)ws; ws += (size_t)Nn * NHID * 4;
  float* xw  = (float*)ws; ws += (size_t)Nn * NHID * 4;
  void*  dis = (void*)ws;  ws += (size_t)Nn * 4;        // int counts, then float rsqrt
  float* xwp = (float*)ws; ws += (size_t)Nn * 4;
  float* sc  = (float*)ws; ws += (size_t)Nn * 4;
  int*   topi    = (int*)ws;   ws += (size_t)NGRAPH * KEEP * 4;
  float* topgate = (float*)ws; ws += (size_t)NGRAPH * KEEP * 4;
  float* gvec = (float*)ws; ws += (size_t)NGRAPH * 768 * 4;
  float* g1   = (float*)ws; ws += (size_t)NGRAPH * NHID * 4;
  float* g2   = (float*)ws; ws += (size_t)NGRAPH * 64 * 4;

  const int BT = 256;
  auto cdiv = [](long long a, long long b) { return (int)((a + b - 1) / b); };

  // degree -> dis
  hipMemsetAsync(dis, 0, (size_t)Nn * 4, stream);
  deg_count_kernel<<<cdiv(E, BT), BT, 0, stream>>>(dst, (int*)dis, E);
  dis_kernel<<<cdiv(Nn, BT), BT, 0, stream>>>(dis, Nn);
  const float* disf = (const float*)dis;

  // GCN layer: hout = relu(agg + dis^2*xw + b)
  auto gcn_layer = [&](const float* hin, int Kd, const float* W, const float* b, float* hout) {
    int tiles = (Nn / 16) * (NHID / 16);
    wmma_gemm_f32_kernel<<<cdiv((long long)tiles * 32, BT), BT, 0, stream>>>(
        hin, W, nullptr, xw, Nn, Kd, NHID, 0);
    hipMemsetAsync(hout, 0, (size_t)Nn * NHID * 4, stream);
    scatter_kernel<<<cdiv((long long)E * 32, BT), BT, 0, stream>>>(src, dst, disf, xw, hout, E);
    combine_kernel<<<cdiv((long long)Nn * NHID, BT), BT, 0, stream>>>(hout, xw, disf, b, Nn);
  };
  gcn_layer(x,  FIN,  W1, b1, h1);
  gcn_layer(h1, NHID, W2, b2, h2);
  gcn_layer(h2, NHID, W3, b3, h3);

  // score GCN (3*NHID -> 1)
  score_linear_kernel<<<cdiv(Nn, BT), BT, 0, stream>>>(h1, h2, h3, Wp, xwp, Nn);
  hipMemsetAsync(sc, 0, (size_t)Nn * 4, stream);
  scatter_score_kernel<<<cdiv(E, BT), BT, 0, stream>>>(src, dst, disf, xwp, sc, E);
  score_combine_kernel<<<cdiv(Nn, BT), BT, 0, stream>>>(sc, xwp, disf, bp, Nn);

  // per-graph top-K + gated pooling
  topk_kernel<<<NGRAPH, NPG, 0, stream>>>(sc, topi, topgate);
  pool_kernel<<<NGRAPH, 3 * NHID, 0, stream>>>(h1, h2, h3, topi, topgate, gvec);

  // MLP head (WMMA for the two big layers)
  {
    int tiles = (NGRAPH / 16) * (NHID / 16);
    wmma_gemm_f32_kernel<<<cdiv((long long)tiles * 32, BT), BT, 0, stream>>>(
        gvec, WL1, bL1, g1, NGRAPH, 6 * NHID, NHID, 1);
  }
  {
    int tiles = (NGRAPH / 16) * (64 / 16);
    wmma_gemm_f32_kernel<<<cdiv((long long)tiles * 32, BT), BT, 0, stream>>>(
        g1, WL2, bL2, g2, NGRAPH, NHID, 64, 1);
  }
  head_kernel<<<cdiv(NGRAPH, BT), BT, 0, stream>>>(g2, WL3, bL3, (float*)d_out);
}